// LearnedCombinatorOp_27144193310732
// MI455X (gfx1250) — compile-verified
//
#include <hip/hip_runtime.h>

typedef __attribute__((ext_vector_type(2))) float v2f;
typedef __attribute__((ext_vector_type(8))) float v8f;

#define WMMA_F32_16x16x4(A, B, C) \
    __builtin_amdgcn_wmma_f32_16x16x4_f32(false, (A), false, (B), (short)0, (C), false, false)

// Problem constants (match reference)
constexpr int Bb = 4, Nn = 2048, MO = 32, Dd = 256, DH = 256;
constexpr int P   = Bb * Nn;   // 8192 positions
constexpr int PT  = 16;        // positions per workgroup
constexpr int NBLK = P / PT;   // 512 workgroups
constexpr int THREADS = 512;   // 16 wave32s

// Dynamic LDS layout (floats):
//   feats : 16*4       =    64
//   h1    : 16*256     =  4096
//   h2    : 16*256     =  4096
//   attn  : 16*16*64   = 16384   (one 16-row chunk of logits/attn per position)
constexpr int LDS_FLOATS = 64 + PT * DH + PT * DH + PT * 16 * 64;

__device__ __forceinline__ float gelu_exact(float x) {
    // torch nn.GELU default: 0.5*x*(1+erf(x/sqrt(2)))
    return 0.5f * x * (1.0f + erff(x * 0.70710678118654752f));
}

__global__ void __launch_bounds__(THREADS)
learned_combinator_kernel(
    const float* __restrict__ left_buf,  const float* __restrict__ left_count,
    const float* __restrict__ right_buf, const float* __restrict__ right_count,
    const int*   __restrict__ subs,
    const float* __restrict__ W1, const float* __restrict__ b1,
    const float* __restrict__ W2, const float* __restrict__ b2,
    const float* __restrict__ W3, const float* __restrict__ b3,
    float* __restrict__ out_buf, float* __restrict__ out_count)
{
    extern __shared__ float smem[];
    float* feats_s = smem;                       // 16 x 4
    float* h1s     = smem + 64;                  // 16 x 256 (pos-major)
    float* h2s     = h1s + PT * DH;              // 16 x 256 (pos-major)
    float* attn_s  = h2s + PT * DH;              // 16 x (16*64) chunk

    const int tid   = threadIdx.x;
    const int lane  = tid & 31;
    const int wave  = tid >> 5;                  // 0..15
    const int pbase = blockIdx.x * PT;

    // Warm caches for this wave's einsum position (global_prefetch_b8 path)
    {
        const size_t pg = (size_t)(pbase + wave);
        __builtin_prefetch(left_buf  + pg * (MO * Dd) + lane * 16, 0, 0);
        __builtin_prefetch(right_buf + pg * (MO * Dd) + lane * 16, 0, 0);
    }

    // ---------------- Phase 0: features + new_count ----------------
    if (tid < PT) {
        const int p = pbase + tid;
        const float lc = left_count[p];
        const float rc = right_count[p];
        const int   s  = subs[p];
        feats_s[tid * 4 + 0] = lc * (1.0f / (float)MO);
        feats_s[tid * 4 + 1] = rc * (1.0f / (float)MO);
        feats_s[tid * 4 + 2] = (s <= 0) ? 1.0f : 0.0f;   // one_hot(clip(s,0,1))[0]
        feats_s[tid * 4 + 3] = (s >= 1) ? 1.0f : 0.0f;   // one_hot(clip(s,0,1))[1]
        out_count[p] = fminf(lc + rc, (float)MO);
    }
    __syncthreads();

    // ---------------- Phase 1: h1 = gelu(feats @ W1^T + b1)  (K=4, VALU) ----
    {
        const int j     = tid & 255;             // hidden unit
        const int phalf = (tid >> 8) * 8;        // positions 0-7 or 8-15
        const float4 w  = *(const float4*)(W1 + j * 4);
        const float bb  = b1[j];
        #pragma unroll
        for (int pp = 0; pp < 8; ++pp) {
            const int p = phalf + pp;
            const float* f = feats_s + p * 4;
            float x = fmaf(f[0], w.x, fmaf(f[1], w.y, fmaf(f[2], w.z, fmaf(f[3], w.w, bb))));
            h1s[p * DH + j] = gelu_exact(x);
        }
    }
    __syncthreads();

    // Lane decomposition shared by all WMMA stages (f32 16x16x4 layouts):
    //   A tile: lane 0-15 => M = lane, K pair {0,1}; lane 16-31 => K pair {2,3}
    //   B tile: lane 0-15 => N = lane, K pair {0,1}; lane 16-31 => K pair {2,3}
    const int mn  = lane & 15;                   // M (A) / N (B) within tile
    const int klo = (lane >> 4) << 1;            // 0 or 2
    const int moff = (lane >> 4) << 3;           // D rows: M = r + moff

    // ---------------- Phase 2: h2 = gelu(h1 @ W2^T + b2)  (WMMA f32) -------
    {
        const int n = wave * 16 + mn;            // one N-tile per wave (16 tiles)
        v8f acc = {};
        #pragma unroll 8
        for (int ks = 0; ks < DH / 4; ++ks) {
            const int k = ks * 4 + klo;
            v2f a = *(const v2f*)(h1s + mn * DH + k);       // A[m][k..k+1]
            v2f b = *(const v2f*)(W2  + (size_t)n * DH + k); // B[k][n] = W2[n][k]
            acc = WMMA_F32_16x16x4(a, b, acc);
        }
        const float bb = b2[n];
        #pragma unroll
        for (int r = 0; r < 8; ++r)
            h2s[(r + moff) * DH + n] = gelu_exact(acc[r] + bb);
    }
    __syncthreads();

    // ---------------- Phase 3: two chunks of 16 attn rows -------------------
    for (int chunk = 0; chunk < 2; ++chunk) {
        // --- logits chunk: n in [chunk*1024, chunk*1024+1024), 64 N-tiles ---
        #pragma unroll
        for (int t = 0; t < 4; ++t) {
            const int ntile = chunk * 64 + wave * 4 + t;
            const int n     = ntile * 16 + mn;
            v8f acc = {};
            #pragma unroll 8
            for (int ks = 0; ks < DH / 4; ++ks) {
                const int k = ks * 4 + klo;
                v2f a = *(const v2f*)(h2s + mn * DH + k);
                v2f b = *(const v2f*)(W3  + (size_t)n * DH + k); // W3[n][k]
                acc = WMMA_F32_16x16x4(a, b, acc);
            }
            const float bb   = b3[n];
            const int   nloc = n - chunk * 1024;       // 0..1023 within chunk
            #pragma unroll
            for (int r = 0; r < 8; ++r)
                attn_s[(r + moff) * 1024 + nloc] = acc[r] + bb;
        }
        __syncthreads();

        // --- softmax over 64-wide rows: 16 pos * 16 rows = 256 rows ---------
        if (tid < 256) {
            float* row = attn_s + (tid >> 4) * 1024 + (tid & 15) * 64;
            float mx = -3.0e38f;
            #pragma unroll 8
            for (int i = 0; i < 64; ++i) mx = fmaxf(mx, row[i]);
            float sum = 0.0f;
            #pragma unroll 8
            for (int i = 0; i < 64; ++i) { float e = __expf(row[i] - mx); row[i] = e; sum += e; }
            const float inv = 1.0f / sum;
            #pragma unroll 8
            for (int i = 0; i < 64; ++i) row[i] *= inv;
        }
        __syncthreads();

        // --- einsum chunk: new_buf[m] = sum_q attn[m][q] * concat[q][:] -----
        // One position per wave; M-tile = 16 rows (this chunk), K = 64, N = 256.
        {
            const int    p  = wave;
            const size_t pg = (size_t)(pbase + p);
            const float* attn_p = attn_s + p * 1024;
            const float* lsrc = left_buf  + pg * (MO * Dd);
            const float* rsrc = right_buf + pg * (MO * Dd);

            #pragma unroll 4
            for (int dt = 0; dt < Dd / 16; ++dt) {
                const int n = dt * 16 + mn;
                v8f acc = {};
                #pragma unroll 4
                for (int ks = 0; ks < 16; ++ks) {
                    const int k = ks * 4 + klo;                 // even, never straddles 32
                    v2f a = *(const v2f*)(attn_p + mn * 64 + k);
                    const float* src = (k < 32) ? (lsrc + (size_t)k * Dd)
                                                : (rsrc + (size_t)(k - 32) * Dd);
                    v2f b;
                    b.x = src[n];        // concat[k][n]
                    b.y = src[Dd + n];   // concat[k+1][n]
                    acc = WMMA_F32_16x16x4(a, b, acc);
                }
                float* outp = out_buf + ((pg * MO) + chunk * 16 + moff) * Dd + n;
                #pragma unroll
                for (int r = 0; r < 8; ++r)
                    outp[(size_t)r * Dd] = acc[r];
            }
        }
        __syncthreads();
    }
}

extern "C" void kernel_launch(void* const* d_in, const int* in_sizes, int n_in,
                              void* d_out, int out_size, void* d_ws, size_t ws_size,
                              hipStream_t stream) {
    const float* left_buf    = (const float*)d_in[0];
    const float* left_count  = (const float*)d_in[1];
    const float* right_buf   = (const float*)d_in[2];
    const float* right_count = (const float*)d_in[3];
    const int*   subs        = (const int*)  d_in[4];
    const float* W1 = (const float*)d_in[5];
    const float* b1 = (const float*)d_in[6];
    const float* W2 = (const float*)d_in[7];
    const float* b2 = (const float*)d_in[8];
    const float* W3 = (const float*)d_in[9];
    const float* b3 = (const float*)d_in[10];

    float* out_buf   = (float*)d_out;                      // (B,N,MO,D)
    float* out_count = (float*)d_out + (size_t)P * MO * Dd; // (B,N)

    const size_t lds_bytes = (size_t)LDS_FLOATS * sizeof(float); // ~98.5 KB of 320 KB/WGP

    learned_combinator_kernel<<<NBLK, THREADS, lds_bytes, stream>>>(
        left_buf, left_count, right_buf, right_count, subs,
        W1, b1, W2, b2, W3, b3, out_buf, out_count);
}